// HeterogenousGraphTransformer_50689204027482
// MI455X (gfx1250) — compile-verified
//
#include <hip/hip_runtime.h>
#include <hip/hip_bf16.h>
#include <math.h>

// Problem constants (match reference)
#define NNODES 100000
#define CCH    128
#define HH     4
#define DD     32
#define EE     200000
#define TT     3

typedef float v2f __attribute__((ext_vector_type(2)));
typedef float v4f __attribute__((ext_vector_type(4)));
typedef float v8f __attribute__((ext_vector_type(8)));

#define LDS_PAD 132   // 128 + 4 floats: bank-conflict-free b64 LDS frag reads

// Force global (addrspace(1)) memory ops even through dynamically-selected
// descriptor pointers (defeats the flat_load fallback seen in round-0 asm).
// Use native ext-vector types: HIP's float4 class can't bind an AS(1) ref.
typedef const __attribute__((address_space(1))) v4f*   gcv4p;
typedef const __attribute__((address_space(1))) float* gcfp;
typedef       __attribute__((address_space(1))) float* gfp;

__device__ __forceinline__ v4f   ld_g4(const float* p) { return *(gcv4p)p; }
__device__ __forceinline__ float ld_g1(const float* p) { return *(gcfp)p; }
__device__ __forceinline__ void  st_g1(float* p, float v) { *(gfp)p = v; }

// ---------------------------------------------------------------------------
// Generic WMMA fp32 GEMM:  Y[M x 128] = concat_k( X[kseg] ) @ W(cols)^T + bias
//   W is row-major [128 x ldw]; K-segment kseg uses columns [kseg*128, +128).
// Block: 256 threads = 8 waves; block tile M=128, N=128; wave tile 16x128.
// Both X tile and W segment staged in LDS (coalesced b128 global loads,
// conflict-free b64 LDS frag reads). 8 v8f accumulators per wave.
// ---------------------------------------------------------------------------
struct GemmDesc {
  const float* X[3];    // K-segment inputs, each [M x 128] row-major
  const float* W;       // [128 x ldw] row-major
  const float* bias;    // [128] or nullptr
  float*       Y;       // [M x 128]
  int          nk;      // number of 128-wide K segments (1 or 3)
  int          ldw;     // row stride of W in floats (128 or 384)
};
struct GemmBatch { GemmDesc d[4]; };

__global__ __launch_bounds__(256)
void hgt_wmma_gemm(GemmBatch batch, int M) {
  const GemmDesc g = batch.d[blockIdx.z];

  __shared__ float Ws[128 * LDS_PAD];
  __shared__ float Xs[128 * LDS_PAD];

  const int tid   = threadIdx.x;
  const int wave  = tid >> 5;
  const int lane  = tid & 31;
  const int lh    = lane >> 4;   // lane half: selects K pair within WMMA frag
  const int ln    = lane & 15;   // M index (A), N index (B/C)
  const int mBase = blockIdx.x * 128;
  const int m0    = mBase + wave * 16;

  v8f acc[8] = {};  // 8 N-tiles of 16x16 fp32 accumulators

  for (int kseg = 0; kseg < g.nk; ++kseg) {
    __syncthreads();
    const float* Wseg = g.W + kseg * 128;
    const float* Xp   = g.X[kseg];
    // Stage W segment (rows n, cols k) and X tile (rows m, cols k) into LDS.
    for (int i = tid; i < 128 * 32; i += 256) {
      const int r = i >> 5;
      const int c = (i & 31) << 2;
      const v4f w4 = ld_g4(Wseg + (size_t)r * g.ldw + c);
      *(v4f*)(&Ws[r * LDS_PAD + c]) = w4;
      const int xr = min(mBase + r, M - 1);  // clamp: dup rows never stored
      const v4f x4 = ld_g4(Xp + (size_t)xr * CCH + c);
      *(v4f*)(&Xs[r * LDS_PAD + c]) = x4;
    }
    __syncthreads();

    const int aoff = (wave * 16 + ln) * LDS_PAD + lh * 2;
    const int boff = ln * LDS_PAD + lh * 2;
    for (int k = 0; k < 128; k += 4) {
      // A frag (16x4 f32): lane ln holds row m0+ln, K pair by lane half
      const v2f A = *(const v2f*)(&Xs[aoff + k]);
#pragma unroll
      for (int nt = 0; nt < 8; ++nt) {
        // B frag (4x16 f32): B[kk][n] = W[n][kk] -> contiguous pair in row n
        const v2f B = *(const v2f*)(&Ws[boff + nt * 16 * LDS_PAD + k]);
        acc[nt] = __builtin_amdgcn_wmma_f32_16x16x4_f32(
            false, A, false, B, (short)0, acc[nt], false, false);
      }
    }
  }

  // Epilogue: C/D layout -> row = m0 + r + 8*lh, col = nt*16 + ln
#pragma unroll
  for (int nt = 0; nt < 8; ++nt) {
    const int col = nt * 16 + ln;
    const float bv = g.bias ? ld_g1(g.bias + col) : 0.0f;
#pragma unroll
    for (int r = 0; r < 8; ++r) {
      const int orow = m0 + lh * 8 + r;
      if (orow < M) st_g1(g.Y + (size_t)orow * CCH + col, acc[nt][r] + bv);
    }
  }
}

// ---------------------------------------------------------------------------
// Attention edge-phase kernels
// ---------------------------------------------------------------------------

// Monotonic float<->uint encoding so atomicMax(uint) == max over floats.
__device__ __forceinline__ unsigned encF(float f) {
  unsigned u = __float_as_uint(f);
  return (u & 0x80000000u) ? ~u : (u | 0x80000000u);
}
__device__ __forceinline__ float decF(unsigned e) {
  return (e & 0x80000000u) ? __uint_as_float(e & 0x7fffffffu)
                           : __uint_as_float(~e);
}

__global__ void hgt_init(unsigned* __restrict__ amaxE,
                         float* __restrict__ denom, int NH) {
  const int i = blockIdx.x * blockDim.x + threadIdx.x;
  if (i < NH) { amaxE[i] = 0u; denom[i] = 0.0f; }
}

// One wave per edge: alpha[e,h] = <q[dst,h,:], k[src,h,:]> / sqrt(D)
__global__ __launch_bounds__(256)
void hgt_attn_logits(const float* __restrict__ Q, const float* __restrict__ K,
                     const int* __restrict__ src, const int* __restrict__ dst,
                     float* __restrict__ alpha, unsigned* __restrict__ amaxE,
                     int E) {
  const int gw   = (blockIdx.x * blockDim.x + threadIdx.x) >> 5;
  const int lane = threadIdx.x & 31;
  if (gw >= E) return;
  const int s = src[gw];
  const int d = dst[gw];
  const v4f q4 = ld_g4(Q + (size_t)d * CCH + lane * 4);
  const v4f k4 = ld_g4(K + (size_t)s * CCH + lane * 4);
  float p = q4.x * k4.x + q4.y * k4.y + q4.z * k4.z + q4.w * k4.w;
  // reduce over the 8 lanes (32 channels) of each head
  p += __shfl_xor(p, 1, 32);
  p += __shfl_xor(p, 2, 32);
  p += __shfl_xor(p, 4, 32);
  if ((lane & 7) == 0) {
    const int h = lane >> 3;
    const float a = p * 0.17677669529663687f;  // 1/sqrt(32)
    alpha[(size_t)gw * HH + h] = a;
    __hip_atomic_fetch_max(&amaxE[(size_t)d * HH + h], encF(a),
                           __ATOMIC_RELAXED, __HIP_MEMORY_SCOPE_AGENT);
  }
}

// Thread per (edge, head): ex = exp(alpha - amax[dst]); denom[dst] += ex
__global__ __launch_bounds__(256)
void hgt_attn_softmax(const int* __restrict__ dst, float* __restrict__ alpha,
                      const unsigned* __restrict__ amaxE,
                      float* __restrict__ denom, int EH) {
  const int i = blockIdx.x * blockDim.x + threadIdx.x;
  if (i >= EH) return;
  const int e = i >> 2;        // HH == 4
  const int h = i & 3;
  const int d = dst[e];
  const float m  = decF(amaxE[(size_t)d * HH + h]);
  const float ex = expf(alpha[i] - m);
  alpha[i] = ex;
  __hip_atomic_fetch_add(&denom[(size_t)d * HH + h], ex,
                         __ATOMIC_RELAXED, __HIP_MEMORY_SCOPE_AGENT);
}

// One wave per edge: O[dst] += (ex / (denom[dst]+1e-16)) * V[src]
__global__ __launch_bounds__(256)
void hgt_attn_agg(const float* __restrict__ V, const int* __restrict__ src,
                  const int* __restrict__ dst, const float* __restrict__ ex,
                  const float* __restrict__ denom, float* __restrict__ O,
                  int E) {
  const int gw   = (blockIdx.x * blockDim.x + threadIdx.x) >> 5;
  const int lane = threadIdx.x & 31;
  if (gw >= E) return;
  const int s = src[gw];
  const int d = dst[gw];
  const int h = lane >> 3;
  const float attn = ex[(size_t)gw * HH + h] /
                     (denom[(size_t)d * HH + h] + 1e-16f);
  const v4f v4 = ld_g4(V + (size_t)s * CCH + lane * 4);
  float* op = O + (size_t)d * CCH + lane * 4;
  __hip_atomic_fetch_add(op + 0, attn * v4.x, __ATOMIC_RELAXED, __HIP_MEMORY_SCOPE_AGENT);
  __hip_atomic_fetch_add(op + 1, attn * v4.y, __ATOMIC_RELAXED, __HIP_MEMORY_SCOPE_AGENT);
  __hip_atomic_fetch_add(op + 2, attn * v4.z, __ATOMIC_RELAXED, __HIP_MEMORY_SCOPE_AGENT);
  __hip_atomic_fetch_add(op + 3, attn * v4.w, __ATOMIC_RELAXED, __HIP_MEMORY_SCOPE_AGENT);
}

// ---------------------------------------------------------------------------
// Host orchestration
// ---------------------------------------------------------------------------
extern "C" void kernel_launch(void* const* d_in, const int* in_sizes, int n_in,
                              void* d_out, int out_size, void* d_ws, size_t ws_size,
                              hipStream_t stream) {
  const float* x   = (const float*)d_in[0];
  const int*   src = (const int*)  d_in[1];
  const int*   dst = (const int*)  d_in[2];
  const float* Wq  = (const float*)d_in[3];
  const float* bq  = (const float*)d_in[4];
  const float* Wk  = (const float*)d_in[5];
  const float* bk  = (const float*)d_in[6];
  const float* Wv  = (const float*)d_in[7];
  const float* bv  = (const float*)d_in[8];
  const float* Wsk = (const float*)d_in[9];
  const float* bs  = (const float*)d_in[10];
  const float* Wf  = (const float*)d_in[11];
  const float* bf  = (const float*)d_in[12];
  float* out = (float*)d_out;

  // Workspace layout (floats): Q,K,V (reused per type), 3x O, alpha, amax, denom
  const size_t NC = (size_t)NNODES * CCH;
  float* Qb = (float*)d_ws;
  float* Kb = Qb + NC;
  float* Vb = Kb + NC;
  float* Ob[TT] = { Vb + NC, Vb + 2 * NC, Vb + 3 * NC };
  float* alphaB = Vb + 4 * NC;                              // EE*HH floats
  unsigned* amaxE = (unsigned*)(alphaB + (size_t)EE * HH);  // NNODES*HH
  float* denom = (float*)(amaxE + (size_t)NNODES * HH);     // NNODES*HH

  const int NH = NNODES * HH;
  const dim3 blk(256);
  const dim3 gemmGrid((NNODES + 127) / 128, 1, 4);
  const int edgeWaveBlocks = (EE * 32) / 256;  // exact: 25000

  for (int t = 0; t < TT; ++t) {
    // Q/K/V/skip projections for this edge type in one batched WMMA launch
    GemmBatch gb{};
    const float* Wmat[4] = { Wq  + (size_t)t * CCH * CCH,
                             Wk  + (size_t)t * CCH * CCH,
                             Wv  + (size_t)t * CCH * CCH,
                             Wsk + (size_t)t * CCH * CCH };
    const float* bvec[4] = { bq + (size_t)t * CCH, bk + (size_t)t * CCH,
                             bv + (size_t)t * CCH, bs + (size_t)t * CCH };
    float* Yp[4] = { Qb, Kb, Vb, Ob[t] };  // Ob[t] seeded with the skip branch
    for (int j = 0; j < 4; ++j) {
      gb.d[j].X[0] = x; gb.d[j].X[1] = x; gb.d[j].X[2] = x;
      gb.d[j].W = Wmat[j]; gb.d[j].bias = bvec[j]; gb.d[j].Y = Yp[j];
      gb.d[j].nk = 1; gb.d[j].ldw = CCH;
    }
    hgt_wmma_gemm<<<gemmGrid, blk, 0, stream>>>(gb, NNODES);

    const int* st = src + (size_t)t * EE;
    const int* dt = dst + (size_t)t * EE;
    hgt_init<<<(NH + 255) / 256, blk, 0, stream>>>(amaxE, denom, NH);
    hgt_attn_logits<<<edgeWaveBlocks, blk, 0, stream>>>(Qb, Kb, st, dt,
                                                        alphaB, amaxE, EE);
    hgt_attn_softmax<<<(EE * HH + 255) / 256, blk, 0, stream>>>(dt, alphaB,
                                                                amaxE, denom,
                                                                EE * HH);
    hgt_attn_agg<<<edgeWaveBlocks, blk, 0, stream>>>(Vb, st, dt, alphaB,
                                                     denom, Ob[t], EE);
  }

  // Fused output GEMM: out = [O0|O1|O2] @ Wf^T + bf  (K = 384, 3 segments)
  GemmBatch gf{};
  gf.d[0].X[0] = Ob[0]; gf.d[0].X[1] = Ob[1]; gf.d[0].X[2] = Ob[2];
  gf.d[0].W = Wf; gf.d[0].bias = bf; gf.d[0].Y = out;
  gf.d[0].nk = TT; gf.d[0].ldw = TT * CCH;
  hgt_wmma_gemm<<<dim3((NNODES + 127) / 128, 1, 1), blk, 0, stream>>>(gf, NNODES);
}